// MemoryOnTpu_87943750353247
// MI455X (gfx1250) — compile-verified
//
#include <hip/hip_runtime.h>

typedef float v2f __attribute__((ext_vector_type(2)));
typedef float v8f __attribute__((ext_vector_type(8)));

#define NQ      256      // queries
#define NDS     16       // datasets
#define NF      128      // feature dim
#define NDB     32768    // database rows per dataset
#define NBKT    128      // num_buckets == num_neighbors
#define BSZ     256      // bucket size (reduction length for argmax)
#define NCHUNK  16       // n-columns handled per workgroup
#define LSTRIDE 132      // padded LDS row stride (floats): 132 % 64 == 4 -> conflict-free B reads
#define MASKVAL -999424.0f  // bf16(-1e6) as f32, matching the reference's bf16 cast
#define VOFF    67108864ull // 256*16*128*128, offset of selected_values in d_out

// ---------------------------------------------------------------------------
// Kernel 1: scores = Q x K^T (f32 WMMA 16x16x4), running argmax over b.
// Double-buffered LDS pipeline: next tile's global loads overlap the current
// tile's 32-deep WMMA chain; one workgroup barrier per iteration.
// grid.x = 2(qhalf) * 16(ds) * 8(n-chunk) = 256 blocks, 256 threads (8 waves)
// wave w handles queries [qhalf*128 + 16w, +16) x d-columns [n0, n0+16)
// ---------------------------------------------------------------------------
__global__ __launch_bounds__(256) void score_argmax_kernel(
    const float* __restrict__ query,   // [256, 16, 128]  (Q, ds, F)
    const float* __restrict__ key_db,  // [16, 32768, 128]
    int* __restrict__ argb)            // [16, 256, 128]  argmax_b per (ds,q,n)
{
  __shared__ float ktile[2][16 * LSTRIDE];
  __shared__ float maskv[2][16];

  const int t    = threadIdx.x;
  const int lane = t & 31;
  const int bx   = blockIdx.x;
  const int nc   = bx & 7;          // n-chunk
  const int ds   = (bx >> 3) & 15;  // dataset
  const int qh   = bx >> 7;         // query half
  const int n0   = nc * NCHUNK;
  const int q0   = qh * 128 + (t >> 5) * 16;  // wave's first query
  const int m    = lane & 15;                 // A row (query) / B col (d) index
  const int kh   = (lane >> 4) * 2;           // K sub-offset per WMMA layout

  // A-matrix: 16 queries x 128 features, resident in registers.
  // Lane layout for f32 16x16x4 A: M = lane&15, VGPR j holds K = 2*(lane>>4)+j.
  v2f a[32];
  const float* qb = query + ((size_t)(q0 + m) * NDS + ds) * NF + kh;
  #pragma unroll
  for (int kk = 0; kk < 32; ++kk)
    a[kk] = *(const v2f*)(qb + kk * 4);

  float best[8];
  int   bi[8];
  #pragma unroll
  for (int r = 0; r < 8; ++r) { best[r] = -3.4e38f; bi[r] = 0; }

  // cooperative key-tile staging: thread t owns 8 floats of row (t>>4)
  const int row = t >> 4;         // 0..15 : local d row
  const int c8  = (t & 15) * 8;   // float column within the 128-wide row
  const float* kbase = key_db + (size_t)ds * NDB * NF;

  // ---- prologue: stage tile for b=0 into buffer 0
  {
    const float* src = kbase + (size_t)(n0 + row) * NF + c8;
    float4 x0 = *(const float4*)(src);
    float4 x1 = *(const float4*)(src + 4);
    float* slot = &ktile[0][row * LSTRIDE + c8];
    *(float4*)(slot)     = x0;
    *(float4*)(slot + 4) = x1;
    float ps = x0.x + x0.y + x0.z + x0.w + x1.x + x1.y + x1.z + x1.w;
    #pragma unroll
    for (int o = 8; o >= 1; o >>= 1) ps += __shfl_xor(ps, o, 16);
    if ((lane & 15) == 0) maskv[0][row] = (ps == 0.0f) ? MASKVAL : 0.0f;
  }
  __syncthreads();

  for (int b = 0; b < BSZ; ++b) {
    const int cur = b & 1;

    // ---- issue next tile's global loads early (overlap with WMMA chain)
    float4 x0, x1;
    const bool have_next = (b + 1 < BSZ);
    if (have_next) {
      const float* src = kbase + (size_t)((b + 1) * NBKT + n0 + row) * NF + c8;
      x0 = *(const float4*)(src);
      x1 = *(const float4*)(src + 4);
    }

    // ---- 16(q) x 16(d) score tile: 32 chained f32 WMMAs over K=128
    // B lane layout mirrors A: N(=local d) = lane&15, VGPR j holds K = 2*(lane>>4)+j
    v8f acc = {};
    const float* bb = &ktile[cur][m * LSTRIDE + kh];
    #pragma unroll
    for (int kk = 0; kk < 32; ++kk) {
      v2f bv = *(const v2f*)(bb + kk * 4);
      acc = __builtin_amdgcn_wmma_f32_16x16x4_f32(
          /*neg_a=*/false, a[kk], /*neg_b=*/false, bv,
          /*c_mod=*/(short)0, acc, /*reuse_a=*/false, /*reuse_b=*/false);
    }

    // ---- running argmax over b (strict > == first-max, like jnp.argmax)
    const float mk = maskv[cur][m];
    #pragma unroll
    for (int r = 0; r < 8; ++r) {
      float s = acc[r] + mk;
      if (s > best[r]) { best[r] = s; bi[r] = b; }
    }

    // ---- commit next tile into the alternate buffer + its zero-row mask
    if (have_next) {
      float* slot = &ktile[1 - cur][row * LSTRIDE + c8];
      *(float4*)(slot)     = x0;
      *(float4*)(slot + 4) = x1;
      float ps = x0.x + x0.y + x0.z + x0.w + x1.x + x1.y + x1.z + x1.w;
      #pragma unroll
      for (int o = 8; o >= 1; o >>= 1) ps += __shfl_xor(ps, o, 16);
      if ((lane & 15) == 0) maskv[1 - cur][row] = (ps == 0.0f) ? MASKVAL : 0.0f;
    }
    __syncthreads();  // next-buffer writes done; prev-buffer reads done
  }

  // C layout: VGPR r, q = q0 + r + 8*(lane>>4), n = n0 + (lane&15)
  const int hi = lane >> 4;
  #pragma unroll
  for (int r = 0; r < 8; ++r) {
    const int qq = q0 + r + 8 * hi;
    argb[((size_t)ds * NQ + qq) * NBKT + n0 + m] = bi[r];
  }
}

// ---------------------------------------------------------------------------
// Kernel 2: gather the winning key/value rows into the [Q, ds, n, F] outputs.
// One wave per (q, ds, n): 2x float4 load + 2x float4 store (512B each row).
// grid.x = 524288/8 = 65536 blocks, 256 threads (8 waves)
// ---------------------------------------------------------------------------
__global__ __launch_bounds__(256) void gather_kernel(
    const float* __restrict__ key_db,
    const float* __restrict__ value_db,
    const int*   __restrict__ argb,
    float* __restrict__ out)
{
  const int t    = threadIdx.x;
  const int lane = t & 31;
  const int wv   = (blockIdx.x << 3) + (t >> 5);  // 0 .. 524287
  const int n    = wv & 127;
  const int ds   = (wv >> 7) & 15;
  const int q    = wv >> 11;

  const int b = argb[((size_t)ds * NQ + q) * NBKT + n];
  const int d = b * NBKT + n;

  const size_t src = ((size_t)ds * NDB + d) * NF + lane * 4;
  const size_t dst = ((((size_t)q * NDS + ds) * NBKT + n) * NF) + lane * 4;

  float4 kx = *(const float4*)(key_db   + src);
  float4 vx = *(const float4*)(value_db + src);
  *(float4*)(out + dst)        = kx;
  *(float4*)(out + VOFF + dst) = vx;
}

// ---------------------------------------------------------------------------
extern "C" void kernel_launch(void* const* d_in, const int* in_sizes, int n_in,
                              void* d_out, int out_size, void* d_ws, size_t ws_size,
                              hipStream_t stream) {
  const float* query    = (const float*)d_in[0];  // [256,16,128]
  const float* key_db   = (const float*)d_in[1];  // [16,32768,128]
  const float* value_db = (const float*)d_in[2];  // [16,32768,128]
  // d_in[3] = num_neighbors (128), fixed by the problem shape
  float* out  = (float*)d_out;
  int*   argb = (int*)d_ws;  // 16*256*128 ints = 2 MB scratch

  score_argmax_kernel<<<256, 256, 0, stream>>>(query, key_db, argb);
  gather_kernel<<<65536, 256, 0, stream>>>(key_db, value_db, argb, out);
}